// Model_67319317397930
// MI455X (gfx1250) — compile-verified
//
#include <hip/hip_runtime.h>

// Edge softmax (dgl.edge_softmax style), MI455X / gfx1250.
// scores: [E, H=8] f32, dst: [E] i32, num_nodes: device scalar i32.
// out[e,h] = exp(s[e,h] - max_{e':dst=d}) / sum_{e':dst=d} exp(...)
//
// Memory-bound: ~0.9GB raw traffic, tables (2 x 3.2MB) L2-resident,
// serpentine pass ordering to reuse the 192MB L2 for the 205MB score stream.

#define HEADS 8

typedef float v4f __attribute__((ext_vector_type(4)));

// ---------------------------------------------------------------------------
// Native-instruction float atomic max: positive floats are order-isomorphic to
// int bits (atomic_max_i32); negative floats are reverse-ordered as uint bits
// (atomic_min_u32). Init value is -inf (0xFF800000). No CAS loop.
__device__ __forceinline__ void atomic_max_f32(float* addr, float v) {
    int bits = __float_as_int(v);
    if (bits >= 0) {
        atomicMax((int*)addr, bits);
    } else {
        atomicMin((unsigned int*)addr, (unsigned int)bits);
    }
}

// ---------------------------------------------------------------------------
// Init the L2-resident reduction tables. num_nodes lives on device (graph
// capture forbids a sync D2H copy), so read it here and grid-stride.
__global__ void ws_init_kernel(int* __restrict__ max_bits,
                               float* __restrict__ sums,
                               const int* __restrict__ num_nodes_p) {
    long total  = (long)(*num_nodes_p) * HEADS;
    long stride = (long)gridDim.x * blockDim.x;
    for (long i = (long)blockIdx.x * blockDim.x + threadIdx.x; i < total; i += stride) {
        max_bits[i] = 0xFF800000;   // -inf as float bits
        sums[i]     = 0.0f;
    }
}

// ---------------------------------------------------------------------------
// Pass 1 (ascending): per-edge 32B vector load, 8 L2 atomics into max table.
__global__ void pass_max_kernel(const v4f* __restrict__ scores,
                                const int* __restrict__ dst,
                                float* __restrict__ maxws,
                                int E) {
    int e = blockIdx.x * blockDim.x + threadIdx.x;
    if (e >= E) return;
    v4f s0 = scores[2 * e + 0];
    v4f s1 = scores[2 * e + 1];
    float* cell = maxws + (long)dst[e] * HEADS;
    atomic_max_f32(cell + 0, s0.x);
    atomic_max_f32(cell + 1, s0.y);
    atomic_max_f32(cell + 2, s0.z);
    atomic_max_f32(cell + 3, s0.w);
    atomic_max_f32(cell + 4, s1.x);
    atomic_max_f32(cell + 5, s1.y);
    atomic_max_f32(cell + 6, s1.z);
    atomic_max_f32(cell + 7, s1.w);
}

// ---------------------------------------------------------------------------
// Pass 2 (descending): reuse the L2 tail left by pass 1. Recompute exp,
// native global_atomic_add_f32 into the L2-resident sum table.
__global__ void pass_sum_kernel(const v4f* __restrict__ scores,
                                const int* __restrict__ dst,
                                const float* __restrict__ maxws,
                                float* __restrict__ sumws,
                                int E) {
    int g = blockIdx.x * blockDim.x + threadIdx.x;
    if (g >= E) return;
    int e = E - 1 - g;                       // serpentine traversal
    v4f s0 = scores[2 * e + 0];
    v4f s1 = scores[2 * e + 1];
    int d  = dst[e];
    const v4f* m = (const v4f*)(maxws + (long)d * HEADS);
    v4f m0 = m[0];
    v4f m1 = m[1];
    float* sc = sumws + (long)d * HEADS;
    atomicAdd(sc + 0, expf(s0.x - m0.x));
    atomicAdd(sc + 1, expf(s0.y - m0.y));
    atomicAdd(sc + 2, expf(s0.z - m0.z));
    atomicAdd(sc + 3, expf(s0.w - m0.w));
    atomicAdd(sc + 4, expf(s1.x - m1.x));
    atomicAdd(sc + 5, expf(s1.y - m1.y));
    atomicAdd(sc + 6, expf(s1.z - m1.z));
    atomicAdd(sc + 7, expf(s1.w - m1.w));
}

// ---------------------------------------------------------------------------
// Pass 3 (ascending): last use of scores/dst -> non-temporal loads; the 205MB
// output is write-once -> non-temporal store (keeps tables + scores in L2).
__global__ void pass_out_kernel(const v4f* __restrict__ scores,
                                const int* __restrict__ dst,
                                const float* __restrict__ maxws,
                                const float* __restrict__ sumws,
                                v4f* __restrict__ out,
                                int E) {
    int e = blockIdx.x * blockDim.x + threadIdx.x;
    if (e >= E) return;
    v4f s0 = __builtin_nontemporal_load(&scores[2 * e + 0]);
    v4f s1 = __builtin_nontemporal_load(&scores[2 * e + 1]);
    int d  = __builtin_nontemporal_load(&dst[e]);
    const v4f* m  = (const v4f*)(maxws + (long)d * HEADS);
    const v4f* sm = (const v4f*)(sumws + (long)d * HEADS);
    v4f m0 = m[0],  m1 = m[1];
    v4f q0 = sm[0], q1 = sm[1];
    v4f o0, o1;
    o0.x = expf(s0.x - m0.x) / q0.x;
    o0.y = expf(s0.y - m0.y) / q0.y;
    o0.z = expf(s0.z - m0.z) / q0.z;
    o0.w = expf(s0.w - m0.w) / q0.w;
    o1.x = expf(s1.x - m1.x) / q1.x;
    o1.y = expf(s1.y - m1.y) / q1.y;
    o1.z = expf(s1.z - m1.z) / q1.z;
    o1.w = expf(s1.w - m1.w) / q1.w;
    __builtin_nontemporal_store(o0, &out[2 * e + 0]);
    __builtin_nontemporal_store(o1, &out[2 * e + 1]);
}

// ---------------------------------------------------------------------------
extern "C" void kernel_launch(void* const* d_in, const int* in_sizes, int n_in,
                              void* d_out, int out_size, void* d_ws, size_t ws_size,
                              hipStream_t stream) {
    const float* scores    = (const float*)d_in[0];   // [E*H] f32
    const int*   dst       = (const int*)d_in[1];     // [E] i32
    const int*   num_nodes = (const int*)d_in[2];     // device scalar

    const int E = in_sizes[1];                        // edge count

    // Workspace split: first half = max table, second half = sum table.
    // Each needs N*H*4 = 3.2MB; both stay resident in the 192MB L2.
    size_t half  = (ws_size / 2) & ~(size_t)255;
    int*   maxbt = (int*)d_ws;
    float* maxws = (float*)d_ws;
    float* sumws = (float*)((char*)d_ws + half);

    const int threads = 256;                          // 8 wave32s per block
    const int blocks  = (E + threads - 1) / threads;

    ws_init_kernel<<<2048, threads, 0, stream>>>(maxbt, sumws, num_nodes);
    pass_max_kernel<<<blocks, threads, 0, stream>>>((const v4f*)scores, dst, maxws, E);
    pass_sum_kernel<<<blocks, threads, 0, stream>>>((const v4f*)scores, dst, maxws, sumws, E);
    pass_out_kernel<<<blocks, threads, 0, stream>>>((const v4f*)scores, dst, maxws, sumws,
                                                    (v4f*)d_out, E);
}